// MILLoss_33715493273915
// MI455X (gfx1250) — compile-verified
//
#include <hip/hip_runtime.h>
#include <hip/hip_bf16.h>
#include <stdint.h>

// MILLoss (mode='min'), B=131072, C=1024, fp32 logits, int32 targets.
// Memory-bound: 512 MB logits stream -> ~22us floor at 23.3 TB/s.
// CDNA5 path: async global->LDS staging (ASYNCcnt) + wave32 shuffle reductions
// + u64 atomic-min keys for exact segment-min with first-index tie-break.

#define IGNORE_INDEX (-1)
constexpr int kC             = 1024;
constexpr int kWavesPerBlock = 8;
constexpr int kThreads       = kWavesPerBlock * 32;
constexpr int kChunks        = kC / 128;   // 8 x (32 lanes x 16B) = 4KB row

typedef int v4i __attribute__((ext_vector_type(4)));
typedef __attribute__((address_space(1))) v4i* gv4i_p;   // global int4*
typedef __attribute__((address_space(3))) v4i* lv4i_p;   // LDS int4*

// ---------- CDNA5 async global->LDS helpers ----------
__device__ __forceinline__ void async_copy_b128(const float* gsrc, float* ldst) {
#if __has_builtin(__builtin_amdgcn_global_load_async_to_lds_b128)
  __builtin_amdgcn_global_load_async_to_lds_b128(
      (gv4i_p)(uintptr_t)gsrc,
      (lv4i_p)ldst,
      /*offset=*/0, /*cpol=*/0);
#else
  unsigned lds_off = (unsigned)(uintptr_t)ldst;   // low 32 bits = LDS offset
  asm volatile("global_load_async_to_lds_b128 %0, %1, off"
               :: "v"(lds_off), "v"(gsrc) : "memory");
#endif
}

__device__ __forceinline__ void wait_asynccnt0() {
#if __has_builtin(__builtin_amdgcn_s_wait_asynccnt)
  __builtin_amdgcn_s_wait_asynccnt(0);
#else
  asm volatile("s_wait_asynccnt 0" ::: "memory");
#endif
  asm volatile("" ::: "memory");   // keep LDS reads below the wait
}

// ---------- kernel 1: init per-label keys ----------
__global__ void init_keys_kernel(unsigned long long* __restrict__ keys, int c) {
  int i = blockIdx.x * blockDim.x + threadIdx.x;
  if (i < c) keys[i] = ~0ull;
}

// ---------- kernel 2: per-row loss + u64 atomic segment-min ----------
__global__ __launch_bounds__(kThreads)
void row_loss_kernel(const float* __restrict__ logits,
                     const int* __restrict__ target,
                     unsigned long long* __restrict__ keys, int B) {
  __shared__ float4 lds4[kWavesPerBlock * (kC / 4)];   // 32 KB

  const int wave = threadIdx.x >> 5;
  const int lane = threadIdx.x & 31;
  const int r    = blockIdx.x * kWavesPerBlock + wave;
  if (r >= B) return;

  const float* rowp  = logits + (size_t)r * kC;        // wave-uniform (SGPR base)
  float4*      lrow4 = &lds4[wave * (kC / 4)];

  // Stage one 4KB row: 8 async b128 copies, 512B each (32 lanes x 16B).
#pragma unroll
  for (int i = 0; i < kChunks; ++i) {
    const int e4 = i * 32 + lane;                      // float4 index in row
    async_copy_b128(rowp + e4 * 4, (float*)(lrow4 + e4));
  }
  wait_asynccnt0();

  // Read row back (ds_load_b128), lane-local then butterfly max.
  float4 v[kChunks];
  float m = -3.402823466e+38f;
#pragma unroll
  for (int i = 0; i < kChunks; ++i) {
    v[i] = lrow4[i * 32 + lane];
    m = fmaxf(m, fmaxf(fmaxf(v[i].x, v[i].y), fmaxf(v[i].z, v[i].w)));
  }
#pragma unroll
  for (int o = 16; o > 0; o >>= 1) m = fmaxf(m, __shfl_xor(m, o, 32));

  // sum of exp(x - m)
  float s = 0.0f;
#pragma unroll
  for (int i = 0; i < kChunks; ++i) {
    s += __expf(v[i].x - m) + __expf(v[i].y - m) +
         __expf(v[i].z - m) + __expf(v[i].w - m);
  }
#pragma unroll
  for (int o = 16; o > 0; o >>= 1) s += __shfl_xor(s, o, 32);

  if (lane == 0) {
    const int  t     = target[r];
    const bool valid = (t != IGNORE_INDEX);
    const int  te    = valid ? t : 0;
    const float xt   = ((const float*)lrow4)[te];
    const float loss = valid ? (m + __logf(s) - xt) : 0.0f;

    // Monotonic float->uint map (total order), then pack with row index so a
    // single u64 atomicMin yields segment-min with first-index tie-breaking.
    const unsigned lb     = __float_as_uint(loss);
    const unsigned mapped = (lb & 0x80000000u) ? ~lb : (lb | 0x80000000u);
    const unsigned long long key =
        ((unsigned long long)mapped << 32) | (unsigned)r;
    atomicMin(&keys[te], key);
  }
}

// ---------- kernel 3: sum per-label minima / #present labels ----------
__global__ void finalize_kernel(const unsigned long long* __restrict__ keys,
                                float* __restrict__ out, int c) {
  float sum = 0.0f;
  int   cnt = 0;
  for (int i = threadIdx.x; i < c; i += blockDim.x) {
    const unsigned long long k = keys[i];
    if (k != ~0ull) {
      const unsigned u  = (unsigned)(k >> 32);
      const unsigned fb = (u & 0x80000000u) ? (u ^ 0x80000000u) : ~u;
      sum += __uint_as_float(fb);
      cnt += 1;
    }
  }
#pragma unroll
  for (int o = 16; o > 0; o >>= 1) {
    sum += __shfl_xor(sum, o, 32);
    cnt += __shfl_xor(cnt, o, 32);
  }
  __shared__ float ssum[8];
  __shared__ int   scnt[8];
  const int wave = threadIdx.x >> 5, lane = threadIdx.x & 31;
  if (lane == 0) { ssum[wave] = sum; scnt[wave] = cnt; }
  __syncthreads();
  if (threadIdx.x == 0) {
    float S = 0.0f; int N = 0;
    for (int w = 0; w < (int)(blockDim.x >> 5); ++w) { S += ssum[w]; N += scnt[w]; }
    out[0] = (N > 0) ? (S / (float)N) : 0.0f;
  }
}

extern "C" void kernel_launch(void* const* d_in, const int* in_sizes, int n_in,
                              void* d_out, int out_size, void* d_ws, size_t ws_size,
                              hipStream_t stream) {
  const float* logits = (const float*)d_in[0];
  const int*   target = (const int*)d_in[1];
  const int B = in_sizes[1];                 // 131072
  (void)n_in; (void)out_size; (void)ws_size;

  unsigned long long* keys = (unsigned long long*)d_ws;   // 8 KB

  init_keys_kernel<<<(kC + 255) / 256, 256, 0, stream>>>(keys, kC);

  const int blocks = (B + kWavesPerBlock - 1) / kWavesPerBlock;
  row_loss_kernel<<<blocks, kThreads, 0, stream>>>(logits, target, keys, B);

  finalize_kernel<<<1, 256, 0, stream>>>(keys, (float*)d_out, kC);
}